// PointerNet_57956288692730
// MI455X (gfx1250) — compile-verified
//
#include <hip/hip_runtime.h>
#include <math.h>

// ---------------------------------------------------------------------------
// Pointer network forward for MI455X (gfx1250), full f32 precision using
// V_WMMA_F32_16X16X4_F32 for every GEMM (embedding, LSTM gates, projections).
// Sequential LSTM is expressed as 512 graph nodes; h is double-buffered.
// ---------------------------------------------------------------------------

#define Bsz   128
#define Ssz   512
#define Hsz   256
#define Esz   128
#define QOSsz 8
#define KCAT  16
#define INW   9     // 1 + QOS
#define KIN   136   // E + QOS
#define TANHC 10.0f

#define NEG_INF (-__builtin_inff())

typedef __attribute__((ext_vector_type(2))) float v2f;
typedef __attribute__((ext_vector_type(8))) float v8f;

__device__ __forceinline__ v8f wmma4(v2f a, v2f b, v8f c) {
  // D = A(16x4) * B(4x16) + C, full f32. Immediates hardcoded at the call.
  return __builtin_amdgcn_wmma_f32_16x16x4_f32(false, a, false, b, (short)0, c,
                                               false, false);
}

// A-fragment (and B-fragment from a row-major W, since we compute x @ W^T):
// lane<16 : rows r0+lane, k = {k0, k0+1};  lane>=16 : k = {k0+2, k0+3}.
__device__ __forceinline__ v2f frag2(const float* __restrict__ base, long ld,
                                     int r0, int k0, int lane) {
  int r = r0 + (lane & 15);
  int k = k0 + ((lane >> 4) << 1);
  const float* p = base + (long)r * ld + k;
  v2f o; o.x = p[0]; o.y = p[1];
  return o;
}

__device__ __forceinline__ float sigmoidf_(float x) {
  return 1.0f / (1.0f + __expf(-x));
}

__device__ __forceinline__ float rng01(unsigned a, unsigned b) {
  unsigned x = a * 0x9E3779B1u ^ (b + 0x7F4A7C15u);
  x ^= x >> 16; x *= 0x85EBCA6Bu; x ^= x >> 13; x *= 0xC2B2AE35u; x ^= x >> 16;
  return ((float)(x >> 8) + 0.5f) * (1.0f / 16777216.0f);
}

// ---------------------------------------------------------------------------
// Generic  Out[M,N] = A[M,K] @ W[N,K]^T + bias   (one 16x16 tile per wave)
// ---------------------------------------------------------------------------
__global__ void k_gemm_wt_bias(const float* __restrict__ A, long lda,
                               const float* __restrict__ W, long ldw,
                               const float* __restrict__ bias,
                               float* __restrict__ Out, long ldo, int K) {
  int lane = threadIdx.x;
  int m0 = blockIdx.x * 16, n0 = blockIdx.y * 16;
  v8f acc = {};
  for (int k0 = 0; k0 < K; k0 += 4) {
    v2f a = frag2(A, lda, m0, k0, lane);
    v2f b = frag2(W, ldw, n0, k0, lane);
    acc = wmma4(a, b, acc);
  }
  int col = n0 + (lane & 15);
  float bb = bias ? bias[col] : 0.0f;
  int rbase = m0 + ((lane >> 4) << 3);
#pragma unroll
  for (int v = 0; v < 8; v++)
    Out[(long)(rbase + v) * ldo + col] = acc[v] + bb;
}

// ---------------------------------------------------------------------------
// Embedding: emb[r,:] = concat(emb1[cat[r]], qos[r]) @ W2^T + b2
// ---------------------------------------------------------------------------
__device__ __forceinline__ float embed_in(const float* __restrict__ inp,
                                          const float* __restrict__ emb1,
                                          int r, int k) {
  if (k < Esz) {
    int c = (int)inp[(long)r * INW];
    return emb1[c * Esz + k];
  }
  return inp[(long)r * INW + 1 + (k - Esz)];
}

__global__ void k_embed(const float* __restrict__ inp,
                        const float* __restrict__ emb1,
                        const float* __restrict__ W2,
                        const float* __restrict__ b2,
                        float* __restrict__ emb) {
  int lane = threadIdx.x;
  int m0 = blockIdx.x * 16, n0 = blockIdx.y * 16;
  v8f acc = {};
  for (int k0 = 0; k0 < KIN; k0 += 4) {
    int r = m0 + (lane & 15);
    int k = k0 + ((lane >> 4) << 1);
    v2f a; a.x = embed_in(inp, emb1, r, k); a.y = embed_in(inp, emb1, r, k + 1);
    v2f b = frag2(W2, KIN, n0, k0, lane);
    acc = wmma4(a, b, acc);
  }
  int col = n0 + (lane & 15);
  float bb = b2[col];
  int rbase = m0 + ((lane >> 4) << 3);
#pragma unroll
  for (int v = 0; v < 8; v++)
    emb[(long)(rbase + v) * Hsz + col] = acc[v] + bb;
}

// ---------------------------------------------------------------------------
// One fused LSTM step: g = X@Wih^T + Hin@Whh^T + bih + bhh ; cell update.
// grid(8,16) x 32 threads: wave (bt,jt) owns 16 batch rows x 16 hidden units
// (computes the 4 corresponding gate tiles, so the update is local).
// ---------------------------------------------------------------------------
__global__ void k_lstm_step(const float* __restrict__ X, long ldx,
                            const float* __restrict__ Hin,
                            const float* __restrict__ Wih,
                            const float* __restrict__ Whh,
                            const float* __restrict__ bih,
                            const float* __restrict__ bhh,
                            float* __restrict__ C,
                            float* __restrict__ Hout,
                            float* __restrict__ seq_out,  // null for decoder
                            long seq_ld) {
  int lane = threadIdx.x;
  int m0 = blockIdx.x * 16;   // batch tile
  int j0 = blockIdx.y * 16;   // hidden-unit tile
  v8f a0 = {}, a1 = {}, a2 = {}, a3 = {};
  for (int k0 = 0; k0 < Hsz; k0 += 4) {
    v2f a = frag2(X, ldx, m0, k0, lane);
    a0 = wmma4(a, frag2(Wih, Hsz, 0 * Hsz + j0, k0, lane), a0);
    a1 = wmma4(a, frag2(Wih, Hsz, 1 * Hsz + j0, k0, lane), a1);
    a2 = wmma4(a, frag2(Wih, Hsz, 2 * Hsz + j0, k0, lane), a2);
    a3 = wmma4(a, frag2(Wih, Hsz, 3 * Hsz + j0, k0, lane), a3);
  }
  for (int k0 = 0; k0 < Hsz; k0 += 4) {
    v2f a = frag2(Hin, Hsz, m0, k0, lane);
    a0 = wmma4(a, frag2(Whh, Hsz, 0 * Hsz + j0, k0, lane), a0);
    a1 = wmma4(a, frag2(Whh, Hsz, 1 * Hsz + j0, k0, lane), a1);
    a2 = wmma4(a, frag2(Whh, Hsz, 2 * Hsz + j0, k0, lane), a2);
    a3 = wmma4(a, frag2(Whh, Hsz, 3 * Hsz + j0, k0, lane), a3);
  }
  int j = j0 + (lane & 15);
  float bi = bih[0 * Hsz + j] + bhh[0 * Hsz + j];
  float bf = bih[1 * Hsz + j] + bhh[1 * Hsz + j];
  float bg = bih[2 * Hsz + j] + bhh[2 * Hsz + j];
  float bo = bih[3 * Hsz + j] + bhh[3 * Hsz + j];
  int rbase = m0 + ((lane >> 4) << 3);
#pragma unroll
  for (int v = 0; v < 8; v++) {
    int b = rbase + v;
    float gi = sigmoidf_(a0[v] + bi);
    float gf = sigmoidf_(a1[v] + bf);
    float gg = tanhf(a2[v] + bg);
    float go = sigmoidf_(a3[v] + bo);
    long ci = (long)b * Hsz + j;
    float c = gf * C[ci] + gi * gg;
    C[ci] = c;
    float h = go * tanhf(c);
    Hout[ci] = h;
    if (seq_out) seq_out[(long)b * seq_ld + j] = h;
  }
}

// ---------------------------------------------------------------------------
// Glimpse logits: lg[b,s] = mask ? -inf : V . tanh(q[b,:] + r[b,s,:])
// ---------------------------------------------------------------------------
__global__ void k_attn_glimpse(const float* __restrict__ q,
                               const float* __restrict__ r,
                               const float* __restrict__ V,
                               const float* __restrict__ mask,
                               float* __restrict__ lg) {
  __shared__ float sq[Hsz], sv[Hsz];
  int b = blockIdx.y;
  int tid = threadIdx.x;
  int s = blockIdx.x * 256 + tid;
  sq[tid] = q[b * Hsz + tid];
  sv[tid] = V[tid];
  __syncthreads();
  const float* rr = r + ((long)b * Ssz + s) * Hsz;
  float acc = 0.0f;
  for (int h = 0; h < Hsz; h++) acc += sv[h] * tanhf(sq[h] + rr[h]);
  long idx = (long)b * Ssz + s;
  lg[idx] = (mask[idx] != 0.0f) ? NEG_INF : acc;
}

// ---------------------------------------------------------------------------
// Pointer logits: lp = 10*tanh(V . tanh(q2+r)); mask -> -inf; store logits;
// ln = in-window ? lp + alpha*latent : -inf
// ---------------------------------------------------------------------------
__global__ void k_attn_pointer(const float* __restrict__ q,
                               const float* __restrict__ r,
                               const float* __restrict__ V,
                               const float* __restrict__ mask,
                               const float* __restrict__ lat_k,
                               const float* __restrict__ alpha_p,
                               int kstep,
                               float* __restrict__ logits_out,
                               float* __restrict__ ln) {
  __shared__ float sq[Hsz], sv[Hsz];
  int b = blockIdx.y;
  int tid = threadIdx.x;
  int s = blockIdx.x * 256 + tid;
  sq[tid] = q[b * Hsz + tid];
  sv[tid] = V[tid];
  __syncthreads();
  const float* rr = r + ((long)b * Ssz + s) * Hsz;
  float acc = 0.0f;
  for (int h = 0; h < Hsz; h++) acc += sv[h] * tanhf(sq[h] + rr[h]);
  float lp = TANHC * tanhf(acc);
  long idx = (long)b * Ssz + s;
  if (mask[idx] != 0.0f) lp = NEG_INF;
  logits_out[idx] = lp;
  bool inwin = (s >> 5) == kstep;  // s / SER_N == k
  ln[idx] = inwin ? lp + alpha_p[0] * lat_k[idx] : NEG_INF;
}

// ---------------------------------------------------------------------------
// Row softmax over S=512 (one block of 256 threads per row)
// ---------------------------------------------------------------------------
__global__ void k_softmax_rows(const float* __restrict__ x,
                               float* __restrict__ p) {
  __shared__ float red[256];
  int b = blockIdx.x, tid = threadIdx.x;
  const float* row = x + (long)b * Ssz;
  red[tid] = fmaxf(row[tid], row[tid + 256]);
  __syncthreads();
  for (int o = 128; o > 0; o >>= 1) {
    if (tid < o) red[tid] = fmaxf(red[tid], red[tid + o]);
    __syncthreads();
  }
  float mx = red[0];
  __syncthreads();
  float e0 = (row[tid] == NEG_INF) ? 0.0f : __expf(row[tid] - mx);
  float e1 = (row[tid + 256] == NEG_INF) ? 0.0f : __expf(row[tid + 256] - mx);
  red[tid] = e0 + e1;
  __syncthreads();
  for (int o = 128; o > 0; o >>= 1) {
    if (tid < o) red[tid] += red[tid + o];
    __syncthreads();
  }
  float inv = 1.0f / red[0];
  p[(long)b * Ssz + tid] = e0 * inv;
  p[(long)b * Ssz + tid + 256] = e1 * inv;
}

// ---------------------------------------------------------------------------
// query[b,h] = sum_s p[b,s] * r[b,s,h]
// ---------------------------------------------------------------------------
__global__ void k_weighted_query(const float* __restrict__ p,
                                 const float* __restrict__ r,
                                 float* __restrict__ query) {
  __shared__ float sp[Ssz];
  int b = blockIdx.x, h = threadIdx.x;
  sp[h] = p[(long)b * Ssz + h];
  sp[h + 256] = p[(long)b * Ssz + h + 256];
  __syncthreads();
  const float* rb = r + (long)b * Ssz * Hsz + h;
  float acc = 0.0f;
  for (int s = 0; s < Ssz; s++) acc += sp[s] * rb[(long)s * Hsz];
  query[b * Hsz + h] = acc;
}

// ---------------------------------------------------------------------------
// Per-row masked/windowed softmax -> probs output; categorical sample via
// deterministic uniform; update mask and dec_in; write idx (as float).
// ---------------------------------------------------------------------------
__global__ void k_sample(const float* __restrict__ ln,
                         const float* __restrict__ emb,
                         float* __restrict__ mask,
                         float* __restrict__ dec_in,
                         float* __restrict__ probs_out,
                         float* __restrict__ idx_out,
                         int kstep) {
  __shared__ float red[256];
  __shared__ int sidx;
  int b = blockIdx.x, tid = threadIdx.x;
  const float* row = ln + (long)b * Ssz;
  red[tid] = fmaxf(row[tid], row[tid + 256]);
  __syncthreads();
  for (int o = 128; o > 0; o >>= 1) {
    if (tid < o) red[tid] = fmaxf(red[tid], red[tid + o]);
    __syncthreads();
  }
  float mx = red[0];
  __syncthreads();
  float e0 = (row[tid] == NEG_INF) ? 0.0f : __expf(row[tid] - mx);
  float e1 = (row[tid + 256] == NEG_INF) ? 0.0f : __expf(row[tid + 256] - mx);
  red[tid] = e0 + e1;
  __syncthreads();
  for (int o = 128; o > 0; o >>= 1) {
    if (tid < o) red[tid] += red[tid + o];
    __syncthreads();
  }
  float ssum = red[0];
  float inv = 1.0f / ssum;
  probs_out[(long)b * Ssz + tid] = e0 * inv;
  probs_out[(long)b * Ssz + tid + 256] = e1 * inv;
  if (tid == 0) {
    float u = rng01((unsigned)kstep * 1315423911u + 7u,
                    (unsigned)b * 2654435761u + 13u);
    float target = u * ssum;
    int base = kstep * 32;  // SER_N window
    float accum = 0.0f;
    int sel = base + 31;
    for (int i = 0; i < 32; i++) {
      int s = base + i;
      float e = (row[s] == NEG_INF) ? 0.0f : __expf(row[s] - mx);
      accum += e;
      if (accum >= target) { sel = s; break; }
    }
    sidx = sel;
    mask[(long)b * Ssz + sel] = 1.0f;
    idx_out[b] = (float)sel;
  }
  __syncthreads();
  dec_in[b * Hsz + tid] = emb[((long)b * Ssz + sidx) * Hsz + tid];
}

// ---------------------------------------------------------------------------
// State init: h double-buffer, c, mask = 0; dec_in = broadcast(dec_start)
// ---------------------------------------------------------------------------
__global__ void k_init(float* __restrict__ hbuf, float* __restrict__ cbuf,
                       float* __restrict__ mask, float* __restrict__ dec_in,
                       const float* __restrict__ dec_start) {
  int i = blockIdx.x * 256 + threadIdx.x;  // 65536 threads
  hbuf[i] = 0.0f;                           // 2*B*H = 65536
  mask[i] = 0.0f;                           // B*S   = 65536
  if (i < Bsz * Hsz) {
    cbuf[i] = 0.0f;
    dec_in[i] = dec_start[i & (Hsz - 1)];
  }
}

// ---------------------------------------------------------------------------
extern "C" void kernel_launch(void* const* d_in, const int* in_sizes, int n_in,
                              void* d_out, int out_size, void* d_ws,
                              size_t ws_size, hipStream_t stream) {
  (void)in_sizes; (void)n_in; (void)out_size; (void)ws_size;
  const float* inputs    = (const float*)d_in[0];
  const float* latent    = (const float*)d_in[1];
  const float* emb1      = (const float*)d_in[2];
  const float* W2        = (const float*)d_in[3];
  const float* b2        = (const float*)d_in[4];
  const float* enc_Wih   = (const float*)d_in[5];
  const float* enc_Whh   = (const float*)d_in[6];
  const float* enc_bih   = (const float*)d_in[7];
  const float* enc_bhh   = (const float*)d_in[8];
  const float* dec_Wih   = (const float*)d_in[9];
  const float* dec_Whh   = (const float*)d_in[10];
  const float* dec_bih   = (const float*)d_in[11];
  const float* dec_bhh   = (const float*)d_in[12];
  const float* ptr_Wq    = (const float*)d_in[13];
  const float* ptr_bq    = (const float*)d_in[14];
  const float* ptr_Wr    = (const float*)d_in[15];
  const float* ptr_br    = (const float*)d_in[16];
  const float* ptr_V     = (const float*)d_in[17];
  const float* glm_Wq    = (const float*)d_in[18];
  const float* glm_bq    = (const float*)d_in[19];
  const float* glm_Wr    = (const float*)d_in[20];
  const float* glm_br    = (const float*)d_in[21];
  const float* glm_V     = (const float*)d_in[22];
  const float* dec_start = (const float*)d_in[23];
  const float* alpha     = (const float*)d_in[24];

  const size_t BSH = (size_t)Bsz * Ssz * Hsz;  // 16.7M floats
  const size_t BH  = (size_t)Bsz * Hsz;
  const size_t BS  = (size_t)Bsz * Ssz;

  float* ws = (float*)d_ws;
  size_t off = 0;
  float* emb     = ws + off; off += BSH;
  float* enc_out = ws + off; off += BSH;
  float* glm_r   = ws + off; off += BSH;
  float* ptr_r   = ws + off; off += BSH;
  float* hbuf    = ws + off; off += 2 * BH;
  float* cbuf    = ws + off; off += BH;
  float* dec_inb = ws + off; off += BH;
  float* maskb   = ws + off; off += BS;
  float* lg      = ws + off; off += BS;
  float* pbuf    = ws + off; off += BS;
  float* lnbuf   = ws + off; off += BS;
  float* query   = ws + off; off += BH;
  float* qtmp    = ws + off; off += BH;

  float* out        = (float*)d_out;
  float* probs_out  = out;                              // [K,B,S]
  float* idx_out    = out + (size_t)KCAT * BS;          // [K,B] as float
  float* logits_out = idx_out + (size_t)KCAT * Bsz;     // [K,B,S]

  // --- init state ---
  k_init<<<dim3(256), dim3(256), 0, stream>>>(hbuf, cbuf, maskb, dec_inb,
                                              dec_start);

  // --- embedding GEMM: [B*S,136] x [136,256] ---
  k_embed<<<dim3(Bsz * Ssz / 16, Hsz / 16), dim3(32), 0, stream>>>(
      inputs, emb1, W2, b2, emb);

  // --- encoder LSTM, 512 sequential steps (h double-buffered) ---
  for (int t = 0; t < Ssz; t++) {
    const float* hin = hbuf + (size_t)(t & 1) * BH;
    float* hout      = hbuf + (size_t)((t + 1) & 1) * BH;
    k_lstm_step<<<dim3(Bsz / 16, Hsz / 16), dim3(32), 0, stream>>>(
        emb + (size_t)t * Hsz, (long)Ssz * Hsz, hin, enc_Wih, enc_Whh,
        enc_bih, enc_bhh, cbuf, hout, enc_out + (size_t)t * Hsz,
        (long)Ssz * Hsz);
  }

  // --- step-invariant reference projections ---
  k_gemm_wt_bias<<<dim3(Bsz * Ssz / 16, Hsz / 16), dim3(32), 0, stream>>>(
      enc_out, Hsz, glm_Wr, Hsz, glm_br, glm_r, Hsz, Hsz);
  k_gemm_wt_bias<<<dim3(Bsz * Ssz / 16, Hsz / 16), dim3(32), 0, stream>>>(
      enc_out, Hsz, ptr_Wr, Hsz, ptr_br, ptr_r, Hsz, Hsz);

  // --- decoder: K_CAT pointer steps ---
  for (int k = 0; k < KCAT; k++) {
    const float* hin = hbuf + (size_t)(k & 1) * BH;
    float* hout      = hbuf + (size_t)((k + 1) & 1) * BH;
    k_lstm_step<<<dim3(Bsz / 16, Hsz / 16), dim3(32), 0, stream>>>(
        dec_inb, (long)Hsz, hin, dec_Wih, dec_Whh, dec_bih, dec_bhh, cbuf,
        hout, nullptr, 0);

    // glimpse: q = h @ Wq^T + bq ; lg ; softmax ; query = sum p*glm_r
    k_gemm_wt_bias<<<dim3(Bsz / 16, Hsz / 16), dim3(32), 0, stream>>>(
        hout, Hsz, glm_Wq, Hsz, glm_bq, qtmp, Hsz, Hsz);
    k_attn_glimpse<<<dim3(Ssz / 256, Bsz), dim3(256), 0, stream>>>(
        qtmp, glm_r, glm_V, maskb, lg);
    k_softmax_rows<<<dim3(Bsz), dim3(256), 0, stream>>>(lg, pbuf);
    k_weighted_query<<<dim3(Bsz), dim3(256), 0, stream>>>(pbuf, glm_r, query);

    // pointer: q2 = query @ Wq^T + bq ; lp -> logits ; ln (window + latent)
    k_gemm_wt_bias<<<dim3(Bsz / 16, Hsz / 16), dim3(32), 0, stream>>>(
        query, Hsz, ptr_Wq, Hsz, ptr_bq, qtmp, Hsz, Hsz);
    k_attn_pointer<<<dim3(Ssz / 256, Bsz), dim3(256), 0, stream>>>(
        qtmp, ptr_r, ptr_V, maskb, latent + (size_t)k * BS, alpha, k,
        logits_out + (size_t)k * BS, lnbuf);

    // softmax -> probs, sample, update mask / dec_in / idx
    k_sample<<<dim3(Bsz), dim3(256), 0, stream>>>(
        lnbuf, emb, maskb, dec_inb, probs_out + (size_t)k * BS,
        idx_out + (size_t)k * Bsz, k);
  }
}